// GIN_net_678604832932
// MI455X (gfx1250) — compile-verified
//
#include <hip/hip_runtime.h>
#include <hip/hip_bf16.h>
#include <math.h>

#define DIM 64
#define IN_DIM 25
#define PROC_STEPS 3

typedef __attribute__((ext_vector_type(2))) float v2f;
typedef __attribute__((ext_vector_type(8))) float v8f;

__device__ __forceinline__ float sigm(float x) { return 1.f / (1.f + expf(-x)); }

// ---------------------------------------------------------------------------
// lin0: h0 = relu(x[N,25] @ W[25,64] + b)   (fp32 WMMA 16x16x4, K padded to 28)
// block = (32,4): 4 waves share one 16-row stripe; wave y owns cols [16y,16y+16)
// A-tile and transposed W staged in LDS -> branch-free ds_load_b64 in K-loop.
// LDS strides (36) chosen so dword bank = (36*l + k) mod 64 is conflict-free.
// ---------------------------------------------------------------------------
#define LSTRIDE0 36   // 28 padded
__global__ __launch_bounds__(128) void k_lin0(const float* __restrict__ x,
                                              const float* __restrict__ W,
                                              const float* __restrict__ bias,
                                              float* __restrict__ h0, int N) {
  const int tid = threadIdx.y * 32 + threadIdx.x;  // 0..127
  const int lane = threadIdx.x;
  const int hi = lane >> 4, l = lane & 15;
  const int row0 = blockIdx.x * 16;
  __shared__ float As[16 * LSTRIDE0];
  __shared__ float Wt[64 * LSTRIDE0];              // Wt[c][k], zero-padded K
  for (int idx = tid; idx < 16 * 28; idx += 128) {
    int r = idx / 28, k = idx - r * 28;
    int row = row0 + r;
    As[r * LSTRIDE0 + k] = (row < N && k < IN_DIM) ? x[row * IN_DIM + k] : 0.f;
  }
  for (int idx = tid; idx < 28 * 64; idx += 128) {
    int k = idx >> 6, cc = idx & 63;               // read W coalesced over cc
    Wt[cc * LSTRIDE0 + k] = (k < IN_DIM) ? W[k * DIM + cc] : 0.f;
  }
  __syncthreads();
  const int col = threadIdx.y * 16 + l;
  v8f c = {};
#pragma unroll
  for (int kk = 0; kk < 28; kk += 4) {
    const int k = kk + hi * 2;                     // f32 A/B layout: VGPR e = K e+2*hi
    v2f a = *(const v2f*)&As[l * LSTRIDE0 + k];
    v2f b = *(const v2f*)&Wt[col * LSTRIDE0 + k];
    c = __builtin_amdgcn_wmma_f32_16x16x4_f32(false, a, false, b, (short)0, c,
                                              false, false);
  }
  const float bv = bias[col];
#pragma unroll
  for (int i = 0; i < 8; ++i) {
    const int r = row0 + hi * 8 + i;               // C/D layout: VGPR i = row i+8*hi
    if (r < N) {
      float v = c[i] + bv;
      h0[r * DIM + col] = v > 0.f ? v : 0.f;
    }
  }
}

// ---------------------------------------------------------------------------
// Edge scatter: aggr[dst,:] += h0[src,:] via HW f32 atomics (4 floats/thread)
// ---------------------------------------------------------------------------
__global__ __launch_bounds__(256) void k_scatter(const long long* __restrict__ ei,
                                                 const float* __restrict__ h0,
                                                 float* __restrict__ aggr,
                                                 long long E) {
  long long t = (long long)blockIdx.x * blockDim.x + threadIdx.x;
  if (t >= E * 16) return;
  long long e = t >> 4;
  int q = (int)(t & 15);
  long long s = ei[e];
  long long d = ei[E + e];
  const float4 v = *(const float4*)(h0 + s * DIM + q * 4);
  float* base = aggr + d * DIM + q * 4;
  unsafeAtomicAdd(base + 0, v.x);
  unsafeAtomicAdd(base + 1, v.y);
  unsafeAtomicAdd(base + 2, v.z);
  unsafeAtomicAdd(base + 3, v.w);
}

// ---------------------------------------------------------------------------
// GIN: h1 = relu((h0 + aggr)[N,64] @ Wg[64,64] + bg)   (fp32 WMMA, K = 64)
// Same LDS-staged structure; A-tile summed during float4 staging.
// ---------------------------------------------------------------------------
#define LSTRIDE1 68   // 64 padded
__global__ __launch_bounds__(128) void k_gin(const float* __restrict__ h0,
                                             const float* __restrict__ aggr,
                                             const float* __restrict__ W,
                                             const float* __restrict__ bias,
                                             float* __restrict__ h1, int N) {
  const int tid = threadIdx.y * 32 + threadIdx.x;
  const int lane = threadIdx.x;
  const int hi = lane >> 4, l = lane & 15;
  const int row0 = blockIdx.x * 16;
  __shared__ float As[16 * LSTRIDE1];
  __shared__ float Wt[64 * LSTRIDE1];
  for (int idx = tid; idx < 16 * 16; idx += 128) { // 16 rows x 16 float4
    int r = idx >> 4, kq = (idx & 15) * 4;
    int row = row0 + r;
    float4 va = {0.f, 0.f, 0.f, 0.f}, vb = {0.f, 0.f, 0.f, 0.f};
    if (row < N) {
      va = *(const float4*)(h0 + (long long)row * DIM + kq);
      vb = *(const float4*)(aggr + (long long)row * DIM + kq);
    }
    float* dst = &As[r * LSTRIDE1 + kq];
    dst[0] = va.x + vb.x; dst[1] = va.y + vb.y;
    dst[2] = va.z + vb.z; dst[3] = va.w + vb.w;
  }
  for (int idx = tid; idx < 64 * 64; idx += 128) {
    int k = idx >> 6, cc = idx & 63;               // coalesced read of W
    Wt[cc * LSTRIDE1 + k] = W[idx];
  }
  __syncthreads();
  const int col = threadIdx.y * 16 + l;
  v8f c = {};
#pragma unroll
  for (int kk = 0; kk < DIM; kk += 4) {
    const int k = kk + hi * 2;
    v2f a = *(const v2f*)&As[l * LSTRIDE1 + k];
    v2f b = *(const v2f*)&Wt[col * LSTRIDE1 + k];
    c = __builtin_amdgcn_wmma_f32_16x16x4_f32(false, a, false, b, (short)0, c,
                                              false, false);
  }
  const float bv = bias[col];
#pragma unroll
  for (int i = 0; i < 8; ++i) {
    const int r = row0 + hi * 8 + i;
    if (r < N) {
      float v = c[i] + bv;
      h1[r * DIM + col] = v > 0.f ? v : 0.f;
    }
  }
}

// ---------------------------------------------------------------------------
// Graph ranges: gptr[g] = lower_bound(batch, g)   (batch is sorted)
// ---------------------------------------------------------------------------
__global__ __launch_bounds__(256) void k_gptr(const long long* __restrict__ batch,
                                              int* __restrict__ gptr, int N, int B) {
  int g = blockIdx.x * blockDim.x + threadIdx.x;
  if (g > B) return;
  int lo = 0, hh = N;
  while (lo < hh) {
    int mid = (lo + hh) >> 1;
    if (batch[mid] < (long long)g) lo = mid + 1; else hh = mid;
  }
  gptr[g] = lo;
}

// ---------------------------------------------------------------------------
// LSTM step: gates = q_star@W_ih^T + h@W_hh^T + b;  (i,f,g,o) cell update
// ---------------------------------------------------------------------------
__global__ __launch_bounds__(256) void k_lstm(const float* __restrict__ q_star,
                                              float* __restrict__ hstate,
                                              float* __restrict__ cstate,
                                              const float* __restrict__ W_ih,
                                              const float* __restrict__ W_hh,
                                              const float* __restrict__ b_ih,
                                              const float* __restrict__ b_hh) {
  const int g = blockIdx.x;
  const int j = threadIdx.x;                 // 0..255 (one gate row each)
  __shared__ float qs[2 * DIM];
  __shared__ float hs[DIM];
  __shared__ float gates[4 * DIM];
  if (j < 2 * DIM) qs[j] = q_star[g * 2 * DIM + j];
  if (j < DIM) hs[j] = hstate[g * DIM + j];
  __syncthreads();
  float acc = b_ih[j] + b_hh[j];
  const float* wi = W_ih + j * (2 * DIM);
#pragma unroll 8
  for (int k = 0; k < 2 * DIM; ++k) acc += qs[k] * wi[k];
  const float* wh = W_hh + j * DIM;
#pragma unroll 8
  for (int k = 0; k < DIM; ++k) acc += hs[k] * wh[k];
  gates[j] = acc;
  __syncthreads();
  if (j < DIM) {
    float ig = gates[j], fg = gates[DIM + j];
    float gg = gates[2 * DIM + j], og = gates[3 * DIM + j];
    float cv = sigm(fg) * cstate[g * DIM + j] + sigm(ig) * tanhf(gg);
    float hv = sigm(og) * tanhf(cv);
    cstate[g * DIM + j] = cv;
    hstate[g * DIM + j] = hv;
  }
}

// ---------------------------------------------------------------------------
// Attention: e = <h1, q[g]>, softmax over graph, r = sum a*h1; q_star=[q, r]
// ---------------------------------------------------------------------------
__global__ __launch_bounds__(128) void k_attn(const float* __restrict__ h1,
                                              const float* __restrict__ hstate,
                                              const int* __restrict__ gptr,
                                              float* __restrict__ ebuf,
                                              float* __restrict__ q_star) {
  const int g = blockIdx.x;
  const int t = threadIdx.x;                 // 0..127
  const int n0 = gptr[g], n1 = gptr[g + 1];
  __shared__ float q[DIM];
  __shared__ float red[128];
  __shared__ float wsh[128];
  if (t < DIM) q[t] = hstate[g * DIM + t];
  __syncthreads();
  float mymax = -INFINITY;
  for (int n = n0 + t; n < n1; n += 128) {
    const float* hr = h1 + (long long)n * DIM;
    float acc = 0.f;
#pragma unroll 8
    for (int k = 0; k < DIM; ++k) acc += hr[k] * q[k];
    ebuf[n] = acc;
    mymax = fmaxf(mymax, acc);
  }
  red[t] = mymax;
  __syncthreads();
  for (int s = 64; s > 0; s >>= 1) {
    if (t < s) red[t] = fmaxf(red[t], red[t + s]);
    __syncthreads();
  }
  const float emax = red[0];
  __syncthreads();
  float rf = 0.f, myden = 0.f;
  for (int base = n0; base < n1; base += 128) {
    int n = base + t;
    float w = (n < n1) ? expf(ebuf[n] - emax) : 0.f;
    wsh[t] = w;
    myden += w;
    __syncthreads();
    if (t < DIM) {
      int cnt = min(128, n1 - base);
      for (int i = 0; i < cnt; ++i)
        rf += wsh[i] * h1[(long long)(base + i) * DIM + t];
    }
    __syncthreads();
  }
  red[t] = myden;
  __syncthreads();
  for (int s = 64; s > 0; s >>= 1) {
    if (t < s) red[t] += red[t + s];
    __syncthreads();
  }
  float denom = red[0];
  if (denom == 0.f) denom = 1.f;
  if (t < DIM) {
    q_star[g * 2 * DIM + t] = q[t];
    q_star[g * 2 * DIM + DIM + t] = rf / denom;
  }
}

// ---------------------------------------------------------------------------
// out[g] = b2 + sum_d lin2_W[d] * relu(q_star[g] @ lin1_W[:,d] + lin1_b[d])
// ---------------------------------------------------------------------------
__global__ __launch_bounds__(64) void k_out(const float* __restrict__ q_star,
                                            const float* __restrict__ W1,
                                            const float* __restrict__ b1,
                                            const float* __restrict__ W2,
                                            const float* __restrict__ b2,
                                            float* __restrict__ out) {
  const int g = blockIdx.x;
  const int d = threadIdx.x;                 // 0..63
  __shared__ float qs[2 * DIM];
  __shared__ float red[DIM];
  qs[d] = q_star[g * 2 * DIM + d];
  qs[DIM + d] = q_star[g * 2 * DIM + DIM + d];
  __syncthreads();
  float acc = b1[d];
#pragma unroll 8
  for (int k = 0; k < 2 * DIM; ++k) acc += qs[k] * W1[k * DIM + d];
  acc = fmaxf(acc, 0.f);
  red[d] = acc * W2[d];
  __syncthreads();
  for (int s = 32; s > 0; s >>= 1) {
    if (d < s) red[d] += red[d + s];
    __syncthreads();
  }
  if (d == 0) out[g] = red[0] + b2[0];
}

// ---------------------------------------------------------------------------
extern "C" void kernel_launch(void* const* d_in, const int* in_sizes, int n_in,
                              void* d_out, int out_size, void* d_ws,
                              size_t ws_size, hipStream_t stream) {
  const float*      x      = (const float*)d_in[0];
  const long long*  ei     = (const long long*)d_in[1];
  const long long*  batch  = (const long long*)d_in[2];
  const float*      lin0_W = (const float*)d_in[3];
  const float*      lin0_b = (const float*)d_in[4];
  const float*      gin_W  = (const float*)d_in[5];
  const float*      gin_b  = (const float*)d_in[6];
  const float*      W_ih   = (const float*)d_in[7];
  const float*      W_hh   = (const float*)d_in[8];
  const float*      b_ih   = (const float*)d_in[9];
  const float*      b_hh   = (const float*)d_in[10];
  const float*      lin1_W = (const float*)d_in[11];
  const float*      lin1_b = (const float*)d_in[12];
  const float*      lin2_W = (const float*)d_in[13];
  const float*      lin2_b = (const float*)d_in[14];
  float*            out    = (float*)d_out;

  const int       N = in_sizes[0] / IN_DIM;
  const long long E = in_sizes[1] / 2;
  const int       B = out_size;               // NUM_GRAPHS

  char* base = (char*)d_ws;
  size_t off = 0;
  auto carve = [&](size_t bytes) {
    char* p = base + off;
    off = (off + bytes + 255) & ~(size_t)255;
    return p;
  };
  float* h0     = (float*)carve((size_t)N * DIM * 4);
  float* aggr   = (float*)carve((size_t)N * DIM * 4);
  float* h1     = (float*)carve((size_t)N * DIM * 4);
  float* ebuf   = (float*)carve((size_t)N * 4);
  int*   gptr   = (int*)carve((size_t)(B + 1) * 4);
  float* hstate = (float*)carve((size_t)B * DIM * 4);
  float* cstate = (float*)carve((size_t)B * DIM * 4);
  float* qstar  = (float*)carve((size_t)B * 2 * DIM * 4);
  (void)ws_size;

  hipMemsetAsync(aggr, 0, (size_t)N * DIM * 4, stream);
  hipMemsetAsync(hstate, 0, (size_t)B * DIM * 4, stream);
  hipMemsetAsync(cstate, 0, (size_t)B * DIM * 4, stream);
  hipMemsetAsync(qstar, 0, (size_t)B * 2 * DIM * 4, stream);

  const int rowTiles = (N + 15) / 16;
  k_lin0<<<dim3(rowTiles), dim3(32, 4), 0, stream>>>(x, lin0_W, lin0_b, h0, N);

  const long long scatThreads = E * 16;
  const int scatBlocks = (int)((scatThreads + 255) / 256);
  k_scatter<<<scatBlocks, 256, 0, stream>>>(ei, h0, aggr, E);

  k_gin<<<dim3(rowTiles), dim3(32, 4), 0, stream>>>(h0, aggr, gin_W, gin_b, h1, N);

  k_gptr<<<(B + 1 + 255) / 256, 256, 0, stream>>>(batch, gptr, N, B);

  for (int step = 0; step < PROC_STEPS; ++step) {
    k_lstm<<<B, 256, 0, stream>>>(qstar, hstate, cstate, W_ih, W_hh, b_ih, b_hh);
    k_attn<<<B, 128, 0, stream>>>(h1, hstate, gptr, ebuf, qstar);
  }

  k_out<<<B, 64, 0, stream>>>(qstar, lin1_W, lin1_b, lin2_W, lin2_b, out);
}